// PAM_32246614458967
// MI455X (gfx1250) — compile-verified
//
#include <hip/hip_runtime.h>
#include <math.h>

// Problem constants (match reference): B=32 batches, F=2048 features.
#define FDIM  2048
#define NBATCH 32
#define BN_EPS 1e-3f

typedef __attribute__((ext_vector_type(2))) float v2f;
typedef __attribute__((ext_vector_type(8))) float v8f;

// Rank-1 outer product of two 16-vectors via V_WMMA_F32_16X16X4_F32.
// A is 16x4 with only K=0 populated (lanes 0-15, VGPR0), B is 4x16 with only
// row K=0 populated (lanes 0-15, VGPR0). D[M,N] = a[M]*b[N] in the standard
// 16x16 f32 C/D layout: lane l, vgpr v -> M = v + 8*(l>>4), N = l&15.
__device__ __forceinline__ v8f outer16(v2f a, v2f b) {
    v8f c = {};
    return __builtin_amdgcn_wmma_f32_16x16x4_f32(
        /*neg_a=*/false, a, /*neg_b=*/false, b,
        /*c_mod=*/(short)0, c, /*reuse_a=*/false, /*reuse_b=*/false);
}

// ---------------------------------------------------------------------------
// K1: b = alpha*x + beta, per-batch max/min of b.  Grid: NBATCH x 256.
// ---------------------------------------------------------------------------
__global__ void pam_prep(const float* __restrict__ x,
                         const float* __restrict__ conv_w,
                         const float* __restrict__ conv_b,
                         const float* __restrict__ bn_gamma,
                         const float* __restrict__ bn_beta,
                         const float* __restrict__ bn_mean,
                         const float* __restrict__ bn_var,
                         float* __restrict__ d_b,
                         float* __restrict__ d_mm) {
    const int batch = blockIdx.x;
    const float scale = bn_gamma[0] * rsqrtf(bn_var[0] + BN_EPS);
    const float alpha = conv_w[0] * scale;
    const float beta  = (conv_b[0] - bn_mean[0]) * scale + bn_beta[0];

    const float4* xr = (const float4*)(x + batch * FDIM);
    float4* br = (float4*)(d_b + batch * FDIM);

    float vmax = -3.4e38f, vmin = 3.4e38f;
    for (int f = threadIdx.x; f < FDIM / 4; f += 256) {
        float4 xv = xr[f];
        float4 bv;
        bv.x = fmaf(alpha, xv.x, beta);
        bv.y = fmaf(alpha, xv.y, beta);
        bv.z = fmaf(alpha, xv.z, beta);
        bv.w = fmaf(alpha, xv.w, beta);
        br[f] = bv;
        vmax = fmaxf(fmaxf(fmaxf(vmax, bv.x), fmaxf(bv.y, bv.z)), bv.w);
        vmin = fminf(fminf(fminf(vmin, bv.x), fminf(bv.y, bv.z)), bv.w);
    }
    // wave32 reduction
    #pragma unroll
    for (int m = 16; m >= 1; m >>= 1) {
        vmax = fmaxf(vmax, __shfl_xor(vmax, m));
        vmin = fminf(vmin, __shfl_xor(vmin, m));
    }
    __shared__ float smax[8], smin[8];
    const int wave = threadIdx.x >> 5, lane = threadIdx.x & 31;
    if (lane == 0) { smax[wave] = vmax; smin[wave] = vmin; }
    __syncthreads();
    if (threadIdx.x == 0) {
        float M = smax[0], mn = smin[0];
        #pragma unroll
        for (int w = 1; w < 8; ++w) { M = fmaxf(M, smax[w]); mn = fminf(mn, smin[w]); }
        d_mm[batch * 2]     = M;
        d_mm[batch * 2 + 1] = mn;
    }
}

// ---------------------------------------------------------------------------
// K2: Z_i = sum_k exp(b_i*b_k - m_i).  Each wave owns 16 rows i, loops over
// all k in 16-wide WMMA tiles.  Grid: NBATCH*16 blocks x 256 threads.
// ---------------------------------------------------------------------------
__global__ void pam_rowsum(const float* __restrict__ d_b,
                           const float* __restrict__ d_mm,
                           float* __restrict__ d_z) {
    const int blocksPerBatch = FDIM / (16 * 8);           // 16
    const int batch = blockIdx.x / blocksPerBatch;
    const int wave  = threadIdx.x >> 5, lane = threadIdx.x & 31;
    const int i0 = ((blockIdx.x % blocksPerBatch) * 8 + wave) * 16;

    __shared__ float s_b[FDIM];                           // 8 KB
    {
        const float4* br = (const float4*)(d_b + batch * FDIM);
        float4* sb4 = (float4*)s_b;
        for (int f = threadIdx.x; f < FDIM / 4; f += 256) sb4[f] = br[f];
    }
    __syncthreads();

    const float bmax = d_mm[batch * 2], bmin = d_mm[batch * 2 + 1];
    const int hi = lane >> 4, l16 = lane & 15;
    const bool lo = (hi == 0);

    // Unconditional LDS load, then select: keeps EXEC untouched (v_cndmask,
    // not an exec-masked ds_load).
    const float aval = s_b[i0 + l16];
    v2f A; A.x = lo ? aval : 0.f; A.y = 0.f;

    float m[8], acc[8];
    #pragma unroll
    for (int v = 0; v < 8; ++v) {
        float bi = s_b[i0 + v + 8 * hi];
        m[v] = (bi >= 0.f) ? bi * bmax : bi * bmin;       // row max of scores
        acc[v] = 0.f;
    }

    for (int k0 = 0; k0 < FDIM; k0 += 16) {
        const float bval = s_b[k0 + l16];
        v2f Bv; Bv.x = lo ? bval : 0.f; Bv.y = 0.f;
        v8f D = outer16(A, Bv);                           // D[M,N] = b_i*b_k
        #pragma unroll
        for (int v = 0; v < 8; ++v)
            acc[v] += __expf(D[v] - m[v]);
    }

    // Row sums: reduce across each 16-lane half (lanes 0-15 hold M=v,
    // lanes 16-31 hold M=v+8).
    #pragma unroll
    for (int v = 0; v < 8; ++v) {
        float a = acc[v];
        a += __shfl_xor(a, 1);
        a += __shfl_xor(a, 2);
        a += __shfl_xor(a, 4);
        a += __shfl_xor(a, 8);
        if (l16 == 0) d_z[batch * FDIM + i0 + v + 8 * hi] = a;
    }
}

// ---------------------------------------------------------------------------
// K3: c[g] = sum_i (b_i/Z_i)*exp(b_i*b_g - m_i); out = gamma*c + x.
// Each wave owns 16 columns g, loops over all i tiles.
// Grid: NBATCH*16 blocks x 256 threads.
// ---------------------------------------------------------------------------
__global__ void pam_out(const float* __restrict__ d_b,
                        const float* __restrict__ d_z,
                        const float* __restrict__ d_mm,
                        const float* __restrict__ x,
                        const float* __restrict__ pam_gamma,
                        float* __restrict__ out) {
    const int blocksPerBatch = FDIM / (16 * 8);           // 16
    const int batch = blockIdx.x / blocksPerBatch;
    const int wave  = threadIdx.x >> 5, lane = threadIdx.x & 31;
    const int g0 = ((blockIdx.x % blocksPerBatch) * 8 + wave) * 16;

    __shared__ float s_b[FDIM], s_w[FDIM], s_m[FDIM];     // 24 KB
    const float bmax = d_mm[batch * 2], bmin = d_mm[batch * 2 + 1];
    {
        const float4* br = (const float4*)(d_b + batch * FDIM);
        const float4* zr = (const float4*)(d_z + batch * FDIM);
        float4* sb4 = (float4*)s_b;
        float4* sw4 = (float4*)s_w;
        float4* sm4 = (float4*)s_m;
        for (int f = threadIdx.x; f < FDIM / 4; f += 256) {
            float4 b = br[f], z = zr[f], w, m;
            w.x = b.x / z.x; w.y = b.y / z.y; w.z = b.z / z.z; w.w = b.w / z.w;
            m.x = (b.x >= 0.f) ? b.x * bmax : b.x * bmin;
            m.y = (b.y >= 0.f) ? b.y * bmax : b.y * bmin;
            m.z = (b.z >= 0.f) ? b.z * bmax : b.z * bmin;
            m.w = (b.w >= 0.f) ? b.w * bmax : b.w * bmin;
            sb4[f] = b; sw4[f] = w; sm4[f] = m;
        }
    }
    __syncthreads();

    const int hi = lane >> 4, l16 = lane & 15;
    const bool lo = (hi == 0);

    // B operand (column values b_g) is loop-invariant.
    const float gval = s_b[g0 + l16];
    v2f Bv; Bv.x = lo ? gval : 0.f; Bv.y = 0.f;

    float cacc = 0.f;
    for (int i0 = 0; i0 < FDIM; i0 += 16) {
        const float aval = s_b[i0 + l16];
        v2f A; A.x = lo ? aval : 0.f; A.y = 0.f;
        v8f D = outer16(A, Bv);                           // D[M,N] = b_i*b_g

        // Each lane-half needs 8 contiguous w/m values: 2x ds_load_b128 each
        // (32B-aligned: i0 is a multiple of 16, 8*hi a multiple of 8).
        const int ibase = i0 + 8 * hi;
        const float4 w0 = *(const float4*)&s_w[ibase];
        const float4 w1 = *(const float4*)&s_w[ibase + 4];
        const float4 m0 = *(const float4*)&s_m[ibase];
        const float4 m1 = *(const float4*)&s_m[ibase + 4];

        cacc = fmaf(w0.x, __expf(D[0] - m0.x), cacc);
        cacc = fmaf(w0.y, __expf(D[1] - m0.y), cacc);
        cacc = fmaf(w0.z, __expf(D[2] - m0.z), cacc);
        cacc = fmaf(w0.w, __expf(D[3] - m0.w), cacc);
        cacc = fmaf(w1.x, __expf(D[4] - m1.x), cacc);
        cacc = fmaf(w1.y, __expf(D[5] - m1.y), cacc);
        cacc = fmaf(w1.z, __expf(D[6] - m1.z), cacc);
        cacc = fmaf(w1.w, __expf(D[7] - m1.w), cacc);
    }
    // Lane l covers rows M=v (hi=0) resp. M=v+8 (hi=1) of column g0+l16:
    // combine the two halves, then lanes 0-15 hold the 16 finished columns.
    cacc += __shfl_xor(cacc, 16);
    if (lo) {
        const int g = g0 + l16;
        out[batch * FDIM + g] = fmaf(cacc, pam_gamma[0], x[batch * FDIM + g]);
    }
}

// ---------------------------------------------------------------------------
extern "C" void kernel_launch(void* const* d_in, const int* in_sizes, int n_in,
                              void* d_out, int out_size, void* d_ws, size_t ws_size,
                              hipStream_t stream) {
    const float* x        = (const float*)d_in[0];
    const float* conv_w   = (const float*)d_in[1];
    const float* conv_b   = (const float*)d_in[2];
    const float* bn_gamma = (const float*)d_in[3];
    const float* bn_beta  = (const float*)d_in[4];
    const float* bn_mean  = (const float*)d_in[5];
    const float* bn_var   = (const float*)d_in[6];
    const float* pg       = (const float*)d_in[7];

    float* d_b  = (float*)d_ws;                 // [NBATCH*FDIM] b values
    float* d_z  = d_b + NBATCH * FDIM;          // [NBATCH*FDIM] row normalizers
    float* d_mm = d_z + NBATCH * FDIM;          // [NBATCH*2] max/min per batch

    pam_prep<<<NBATCH, 256, 0, stream>>>(x, conv_w, conv_b, bn_gamma, bn_beta,
                                         bn_mean, bn_var, d_b, d_mm);
    pam_rowsum<<<NBATCH * 16, 256, 0, stream>>>(d_b, d_mm, d_z);
    pam_out<<<NBATCH * 16, 256, 0, stream>>>(d_b, d_z, d_mm, x, pg, (float*)d_out);
}